// CondFilterT_33921651704221
// MI455X (gfx1250) — compile-verified
//
#include <hip/hip_runtime.h>

// Problem constants from the reference
#define EMB        64
#define NC         50           // conditions per row
#define NIDX       51           // 1 event + 50 conditions
#define ROWSTRIDE  68           // padded LDS row stride in floats: 272B = 17*16 (16B aligned, conflict-free)
#define OUTPER     3264         // 64 + 50*64 floats per batch row

typedef __attribute__((ext_vector_type(2))) float v2f;
typedef __attribute__((ext_vector_type(4))) float v4f;
typedef __attribute__((ext_vector_type(8))) float v8f;

__global__ __launch_bounds__(128) void CondFilterT_kernel(
    const int*   __restrict__ inp,     // [B, 51] int32 indices
    const float* __restrict__ table,   // [NEVENTS+2, 64] f32
    float*       __restrict__ out)     // [B, 3264] f32
{
    // LDS rows: 0..49 = cond embeddings, 50 = event embedding, 51 = zero dummy
    __shared__ float rows[52 * ROWSTRIDE];
    __shared__ float gram[4 * 256];    // one 16x16 f32 Gram tile per wave
    __shared__ int   idx[NIDX];

    const int b    = blockIdx.x;
    const int tid  = threadIdx.x;
    const int lane = tid & 31;
    const int wave = tid >> 5;

    // ---- Stage 1: indices + zero dummy row ----
    if (tid < NIDX) idx[tid] = inp[b * NIDX + tid];
    if (tid < ROWSTRIDE) rows[51 * ROWSTRIDE + tid] = 0.0f;
    __syncthreads();

    // ---- Stage 2: gather 51 rows x 64 floats from the table (L2-resident) as float4 ----
    for (int q = tid; q < NIDX * 16; q += 128) {
        int r      = q >> 4;                       // input slot: 0 = event, 1.. = conds
        int f4     = q & 15;
        int ldsrow = (r == 0) ? 50 : (r - 1);
        const float* src = table + (size_t)idx[r] * EMB + f4 * 4;
        v4f v = *(const v4f*)src;
        *(v4f*)&rows[ldsrow * ROWSTRIDE + f4 * 4] = v;
    }
    __syncthreads();

    // ---- Stage 3: per-wave 16x16 Gram tile via chained V_WMMA_F32_16X16X4_F32 ----
    // Tile rows: m=0 -> event, m=1..15 -> cond (15*wave + m - 1), dummy zeros past 49.
    {
        int m     = lane & 15;
        int condi = wave * 15 + m - 1;
        int r     = (m == 0) ? 50 : ((condi < NC) ? condi : 51);
        // 32-bit A 16x4 layout: lanes 0-15 carry K=(4s+0,4s+1); lanes 16-31 carry K=(4s+2,4s+3)
        const float* rp = &rows[r * ROWSTRIDE + ((lane >= 16) ? 2 : 0)];

        v8f acc = {0.f, 0.f, 0.f, 0.f, 0.f, 0.f, 0.f, 0.f};
        #pragma unroll
        for (int s = 0; s < 16; ++s) {
            v2f a = *(const v2f*)(rp + 4 * s);     // ds_load_b64, 8B aligned
            // Symmetric Gram: B = A^T of the same tile -> identical register fragment.
            acc = __builtin_amdgcn_wmma_f32_16x16x4_f32(
                false, a, false, a, (short)0, acc, false, false);
        }

        // Dump tile to LDS. C/D layout: VGPR v -> M = v (lanes 0-15) or 8+v (lanes 16-31), N = lane%16.
        int n     = lane & 15;
        int mbase = (lane < 16) ? 0 : 8;
        #pragma unroll
        for (int v = 0; v < 8; ++v)
            gram[wave * 256 + (mbase + v) * 16 + n] = acc[v];
    }
    __syncthreads();

    // ---- Stage 4: outputs ----
    // 4a: raw event embedding (first 64 floats)
    const float g00        = gram[0];        // event sum-of-squares (tile 0, (0,0))
    const float inv_norm_e = rsqrtf(g00);
    if (tid < EMB)
        out[(size_t)b * OUTPER + tid] = rows[50 * ROWSTRIDE + tid];

    // 4b: filtered conditions: raw * G0c * rsqrt(G00) / Gcc, vectorized float4 stores
    for (int q = tid; q < NC * 16; q += 128) {
        int c  = q >> 4;
        int f4 = q & 15;
        int w  = c / 15;
        int m  = c % 15 + 1;
        float gcc  = gram[w * 256 + m * 16 + m];
        float g0c  = gram[w * 256 + m];
        float coef = g0c * inv_norm_e / gcc;

        v4f v = *(const v4f*)&rows[c * ROWSTRIDE + f4 * 4];
        v4f o;
        o.x = v.x * coef; o.y = v.y * coef; o.z = v.z * coef; o.w = v.w * coef;
        *(v4f*)&out[(size_t)b * OUTPER + EMB + c * EMB + f4 * 4] = o;
    }
}

extern "C" void kernel_launch(void* const* d_in, const int* in_sizes, int n_in,
                              void* d_out, int out_size, void* d_ws, size_t ws_size,
                              hipStream_t stream) {
    const int*   inp   = (const int*)d_in[0];     // [B, 51] int32
    const float* table = (const float*)d_in[1];   // [NEVENTS+2, 64] f32
    float*       out   = (float*)d_out;           // [B, 3264] f32

    const int B = in_sizes[0] / NIDX;             // 16384
    CondFilterT_kernel<<<B, 128, 0, stream>>>(inp, table, out);
}